// LUAffineFlowModel_71270687310601
// MI455X (gfx1250) — compile-verified
//
#include <hip/hip_runtime.h>
#include <hip/hip_bf16.h>
#include <stdint.h>

// ---------------------------------------------------------------------------
// LU-affine / sigmoid flow, mean log-prob.  MI455X (gfx1250, wave32).
// Transcendental-bound (exp/log/rcp); memory traffic is only 8 MB.
// CDNA5 paths used:
//   * v_wmma_f32_16x16x4_f32 for the one-time W_i = L_i*U_i products
//     (two 2x2 layer products packed block-diagonally per WMMA tile)
//   * global_load_async_to_lds_b64 + s_wait_asynccnt parameter staging
//   * broadcast ds_load_b128 constant reads, wave32 block sizing
// ---------------------------------------------------------------------------

#define NLMAX 64
#define TPB   256          // 8 wave32 waves per block
#define EPT   4            // batch elements per thread (4-way ILP in layer loop)

typedef float v2f __attribute__((ext_vector_type(2)));
typedef float v8f __attribute__((ext_vector_type(8)));

__device__ __forceinline__ void layer_step(float w00, float w01, float w10, float w11,
                                           float b0, float b1,
                                           float& v0, float& v1, float& ld) {
  // u = W v + b
  float u0 = fmaf(w00, v0, fmaf(w01, v1, b0));
  float u1 = fmaf(w10, v0, fmaf(w11, v1, b1));
  // stable: log sigmoid'(u) = -|u| - 2*log(1 + exp(-|u|)); sigmoid via rcp
  float a0 = -fabsf(u0), a1 = -fabsf(u1);
  float e0 = __expf(a0),  e1 = __expf(a1);   // v_exp_f32
  float p0 = 1.0f + e0,   p1 = 1.0f + e1;
  float s0 = __builtin_amdgcn_rcpf(p0);      // v_rcp_f32
  float s1 = __builtin_amdgcn_rcpf(p1);
  ld += (a0 + a1) - 2.0f * (__logf(p0) + __logf(p1));  // v_log_f32
  v0 = (u0 >= 0.0f) ? s0 : e0 * s0;          // sigmoid(u)
  v1 = (u1 >= 0.0f) ? s1 : e1 * s1;
}

// --- kernel 1: W_i = L_i U_i via WMMA (2 layers / 16x16x4 tile), ------------
//     bias copy, logdet = sum log|diag|
__global__ __launch_bounds__(64) void lu_precompute(
    const float* __restrict__ lower, const float* __restrict__ upper,
    const float* __restrict__ diag,  const float* __restrict__ bias,
    float* __restrict__ cparams, int nl) {
  __shared__ float red[NLMAX];
  const int i = threadIdx.x;

  // bias + padding + per-layer logdet terms
  if (i < nl) {
    cparams[i * 8 + 4] = bias[i * 2 + 0];
    cparams[i * 8 + 5] = bias[i * 2 + 1];
    cparams[i * 8 + 6] = 0.0f;
    cparams[i * 8 + 7] = 0.0f;
    red[i] = logf(fabsf(diag[i * 2 + 0])) + logf(fabsf(diag[i * 2 + 1]));
  }

  // W products on the matrix core: wave 0 only (EXEC must be all-ones).
  // A = blockdiag(L_{2t}, L_{2t+1})  (16x4, block rows 0..1 / 2..3, K 0..1 / 2..3)
  // B = blockdiag(U_{2t}, U_{2t+1})  (4x16)
  // D = blockdiag(W_{2t}, W_{2t+1})  (16x16)
  if (i < 32) {
    const int lane = i;
    const int h  = lane >> 4;     // K-band half: lanes 0-15 -> K{0,1}, 16-31 -> K{2,3}
    const int mn = lane & 15;     // row (A) / col (B)
    for (int t = 0; t < (nl + 1) / 2; ++t) {
      const int p = 2 * t + h;    // layer served by this lane's K band
      v2f a = {0.0f, 0.0f};       // A[mn][2h], A[mn][2h+1]
      v2f b = {0.0f, 0.0f};       // B[2h][mn], B[2h+1][mn]
      if (p < nl) {
        const int r = mn - 2 * h; // row/col within the 2x2 block of layer p
        if (r == 0) {             // L row 0 = [1, 0];       U col 0 = [d0, 0]
          a.x = 1.0f;             a.y = 0.0f;
          b.x = diag[p * 2 + 0];  b.y = 0.0f;
        } else if (r == 1) {      // L row 1 = [l10, 1];     U col 1 = [u01, d1]
          a.x = lower[p * 4 + 2]; a.y = 1.0f;
          b.x = upper[p * 4 + 1]; b.y = diag[p * 2 + 1];
        }
      }
      v8f cz = {};
      v8f d = __builtin_amdgcn_wmma_f32_16x16x4_f32(
          false, a, false, b, (short)0, cz, false, false);
      // D layout: lane = N (+16 if M>=8), VGPR = M%8.
      // W_{2t}[m][n]   -> lane n,   VGPR m   (m,n in 0..1)
      // W_{2t+1}[m][n] -> lane n+2, VGPR m+2
      const int q0 = 2 * t, q1 = 2 * t + 1;
      if (lane < 2) {
        cparams[q0 * 8 + 0 + lane] = d[0];          // w0?[0][lane]
        cparams[q0 * 8 + 2 + lane] = d[1];          // w0?[1][lane]
      } else if (lane < 4 && q1 < nl) {
        cparams[q1 * 8 + 0 + (lane - 2)] = d[2];    // w1?[0][lane-2]
        cparams[q1 * 8 + 2 + (lane - 2)] = d[3];    // w1?[1][lane-2]
      }
    }
  }

  __syncthreads();
  if (i == 0) {
    double s = 0.0;
    for (int j = 0; j < nl; ++j) s += (double)red[j];
    cparams[NLMAX * 8] = (float)s;             // logdet_affine
  }
}

// --- kernel 2: main flow, per-block partial sums of (log_base + ld) ---------
template <int NLT>
__global__ __launch_bounds__(TPB) void flow_main(
    const float* __restrict__ x, const float* __restrict__ cparams,
    double* __restrict__ partials, int batch, int nl_rt) {
  const int nl = NLT ? NLT : nl_rt;
  __shared__ __align__(16) float cbuf[NLMAX * 8];   // 2 KB constant table
  __shared__ double sred[TPB];
  const int tid = threadIdx.x;

  // Stage the parameter table global -> LDS with CDNA5 async copies.
  if (tid * 2 < nl * 8) {
    unsigned long long g = (unsigned long long)(uintptr_t)(cparams + tid * 2);
    unsigned int       l = (unsigned int)(uintptr_t)(cbuf + tid * 2);
    asm volatile("global_load_async_to_lds_b64 %0, %1, off"
                 :: "v"(l), "v"(g) : "memory");
  }
  asm volatile("s_wait_asynccnt 0x0" ::: "memory");
  __syncthreads();

  const long long base = (long long)blockIdx.x * (TPB * EPT) + tid;

  long long idx[EPT];
  bool      msk[EPT];
  float v0[EPT], v1[EPT], ld[EPT];
#pragma unroll
  for (int j = 0; j < EPT; ++j) {
    idx[j] = base + (long long)j * TPB;
    msk[j] = idx[j] < (long long)batch;
    long long c = msk[j] ? idx[j] : 0;        // clamp: keep loads in-bounds
    float2 xv = ((const float2*)x)[c];
    v0[j] = xv.x; v1[j] = xv.y; ld[j] = 0.0f;
  }

  // 63 sigmoid layers; constants broadcast from LDS once per layer for all 4
  // interleaved elements (4-way ILP hides v_exp/v_log latency).
  for (int i = 0; i < nl - 1; ++i) {
    float4 w = *((const float4*)(cbuf + i * 8));       // ds_load_b128
    float2 b = *((const float2*)(cbuf + i * 8 + 4));   // ds_load_b64
#pragma unroll
    for (int j = 0; j < EPT; ++j)
      layer_step(w.x, w.y, w.z, w.w, b.x, b.y, v0[j], v1[j], ld[j]);
  }

  // final affine (no sigmoid) + Gaussian base log-density
  const float LOG2PI = 1.8378770664093453f;
  float4 w = *((const float4*)(cbuf + (nl - 1) * 8));
  float2 b = *((const float2*)(cbuf + (nl - 1) * 8 + 4));
  float acc = 0.0f;
#pragma unroll
  for (int j = 0; j < EPT; ++j) {
    float z0 = fmaf(w.x, v0[j], fmaf(w.y, v1[j], b.x));
    float z1 = fmaf(w.z, v0[j], fmaf(w.w, v1[j], b.y));
    float lp = fmaf(-0.5f, fmaf(z0, z0, z1 * z1), -LOG2PI) + ld[j];
    acc += msk[j] ? lp : 0.0f;
  }

  // deterministic block tree-reduce in double
  sred[tid] = (double)acc;
  __syncthreads();
#pragma unroll
  for (int s = TPB / 2; s > 0; s >>= 1) {
    if (tid < s) sred[tid] += sred[tid + s];
    __syncthreads();
  }
  if (tid == 0) partials[blockIdx.x] = sred[0];
}

// --- kernel 3: deterministic final reduction -> mean + logdet ---------------
__global__ __launch_bounds__(TPB) void flow_finalize(
    const double* __restrict__ partials, int nblk,
    const float* __restrict__ cparams, float* __restrict__ out, int batch) {
  __shared__ double sred[TPB];
  int tid = threadIdx.x;
  double s = 0.0;
  for (int i = tid; i < nblk; i += TPB) s += partials[i];
  sred[tid] = s;
  __syncthreads();
#pragma unroll
  for (int k = TPB / 2; k > 0; k >>= 1) {
    if (tid < k) sred[tid] += sred[tid + k];
    __syncthreads();
  }
  if (tid == 0)
    out[0] = (float)(sred[0] / (double)batch + (double)cparams[NLMAX * 8]);
}

// ---------------------------------------------------------------------------
extern "C" void kernel_launch(void* const* d_in, const int* in_sizes, int n_in,
                              void* d_out, int out_size, void* d_ws, size_t ws_size,
                              hipStream_t stream) {
  const float* x     = (const float*)d_in[0];   // [B, 2]
  const float* lower = (const float*)d_in[1];   // [n, 2, 2]
  const float* upper = (const float*)d_in[2];   // [n, 2, 2]
  const float* diag  = (const float*)d_in[3];   // [n, 2]
  const float* bias  = (const float*)d_in[4];   // [n, 2]

  int batch = in_sizes[0] / 2;
  int nl    = in_sizes[3] / 2;
  if (nl > NLMAX) nl = NLMAX;

  float*  cparams  = (float*)d_ws;                       // 513 floats
  double* partials = (double*)((char*)d_ws + 4096);      // per-block sums

  const int per_block = TPB * EPT;                       // 1024 elems/block
  int nblk = (batch + per_block - 1) / per_block;        // 1024 for B=2^20

  lu_precompute<<<1, 64, 0, stream>>>(lower, upper, diag, bias, cparams, nl);
  if (nl == NLMAX)
    flow_main<NLMAX><<<nblk, TPB, 0, stream>>>(x, cparams, partials, batch, nl);
  else
    flow_main<0><<<nblk, TPB, 0, stream>>>(x, cparams, partials, batch, nl);
  flow_finalize<<<1, TPB, 0, stream>>>(partials, nblk, cparams, (float*)d_out, batch);
}